// RSCNN_MSG_SIAMESE_40570261078672
// MI455X (gfx1250) — compile-verified
//
#include <hip/hip_runtime.h>
#include <hip/hip_bf16.h>

// ---------------------------------------------------------------------------
// RSCNN-MSG siamese forward for MI455X (gfx1250, wave32, WMMA f32 16x16x4)
// ---------------------------------------------------------------------------

typedef float v2f __attribute__((ext_vector_type(2)));
typedef float v8f __attribute__((ext_vector_type(8)));

#define BN_SCALE 0.9999950000374997f /* 1/sqrt(1+1e-5) */

static __device__ __forceinline__ float relu_f(float x) { return fmaxf(x, 0.0f); }

// ---------------------------------------------------------------------------
// Furthest point sampling: one block per batch. dists lives in LDS.
// Matches jax scan: idx[0]=0; at step t emit `last`, then last = argmax(dists).
// ---------------------------------------------------------------------------
__global__ __launch_bounds__(256) void fps_kernel(const float* __restrict__ xyz,
                                                  int N, int npoint,
                                                  int* __restrict__ fidx)
{
    const int b   = blockIdx.x;
    const int tid = threadIdx.x;
    __shared__ float dists[4096];
    __shared__ float rv[256];
    __shared__ int   ri[256];
    __shared__ int   s_last;
    if (tid == 0) s_last = 0;
    __syncthreads();

    const float* bx = xyz + (size_t)b * N * 3;

    for (int t = 0; t < npoint; ++t) {
        int last = s_last;
        if (tid == 0) fidx[(size_t)b * npoint + t] = last;
        float px = bx[last * 3 + 0];
        float py = bx[last * 3 + 1];
        float pz = bx[last * 3 + 2];

        float bestv = -1.0f;
        int   besti = 0;
        for (int i = tid; i < N; i += 256) {
            float dx = bx[i * 3 + 0] - px;
            float dy = bx[i * 3 + 1] - py;
            float dz = bx[i * 3 + 2] - pz;
            float d  = dx * dx + dy * dy + dz * dz;
            float nd = (t == 0) ? d : fminf(dists[i], d);
            dists[i] = nd;
            if (nd > bestv) { bestv = nd; besti = i; }
        }
        rv[tid] = bestv;
        ri[tid] = besti;
        __syncthreads();
        // tree argmax; ties -> smaller index (matches jnp.argmax first-max)
        for (int off = 128; off > 0; off >>= 1) {
            if (tid < off) {
                float v2 = rv[tid + off];
                int   i2 = ri[tid + off];
                if (v2 > rv[tid] || (v2 == rv[tid] && i2 < ri[tid])) {
                    rv[tid] = v2;
                    ri[tid] = i2;
                }
            }
            __syncthreads();
        }
        if (tid == 0) s_last = ri[0];
        __syncthreads();
    }
}

// ---------------------------------------------------------------------------
// new_xyz[b,s,:] = xyz[b, fidx[b,s], :]
// ---------------------------------------------------------------------------
__global__ __launch_bounds__(256) void gather_xyz_kernel(const float* __restrict__ xyz,
                                                         const int* __restrict__ fidx,
                                                         float* __restrict__ nxyz,
                                                         int N, int S)
{
    int e = blockIdx.x * 256 + threadIdx.x;
    if (e >= 8 * S) return;
    int b = e / S;
    int p = fidx[e];
    const float* src = xyz + ((size_t)b * N + p) * 3;
    nxyz[(size_t)e * 3 + 0] = src[0];
    nxyz[(size_t)e * 3 + 1] = src[1];
    nxyz[(size_t)e * 3 + 2] = src[2];
}

// ---------------------------------------------------------------------------
// Ball query: thread per (b,s). First `ns` indices within radius (ascending),
// padded with first hit; zero if no hits. Matches the sorted-key reference.
// ---------------------------------------------------------------------------
__global__ __launch_bounds__(256) void ball_query_kernel(const float* __restrict__ xyz,
                                                         const float* __restrict__ nxyz,
                                                         int* __restrict__ idxo,
                                                         int N, int S, int ns, float r2)
{
    int e = blockIdx.x * 256 + threadIdx.x;
    if (e >= 8 * S) return;
    int b = e / S;
    const float* bx = xyz + (size_t)b * N * 3;
    float px = nxyz[(size_t)e * 3 + 0];
    float py = nxyz[(size_t)e * 3 + 1];
    float pz = nxyz[(size_t)e * 3 + 2];
    int cnt = 0, first = -1;
    for (int i = 0; i < N; ++i) {
        float dx = bx[i * 3 + 0] - px;
        float dy = bx[i * 3 + 1] - py;
        float dz = bx[i * 3 + 2] - pz;
        if (dx * dx + dy * dy + dz * dz < r2) {
            if (first < 0) first = i;
            idxo[(size_t)e * ns + cnt] = i;
            if (++cnt == ns) break;
        }
    }
    int pad = (first < 0) ? 0 : first;
    for (int j = cnt; j < ns; ++j) idxo[(size_t)e * ns + j] = pad;
}

// ---------------------------------------------------------------------------
// Fused group -> [ns,Cin]x[Cin,Cout] GEMM (f32 WMMA 16x16x4) -> BN+ReLU ->
// maxpool over ns.  One block per (b,s); 8 waves each own one 16-col N-tile
// and iterate all MT row tiles.  X staged in LDS in K-chunks of KC.
// grid = (S, B, ceil(Cout/128)), block = 256.
// ---------------------------------------------------------------------------
template <int MT, int KC>
__global__ __launch_bounds__(256) void sa_conv_kernel(
    const float* __restrict__ xyz,    // [B,N,3]   (used when useXyz)
    const float* __restrict__ feats,  // [B,N,Cfeat] (may be null if Cfeat==0)
    const int*   __restrict__ idx,    // [B,S,ns]
    const float* __restrict__ nxyz,   // [B,S,3]   (used when useXyz)
    const float* __restrict__ W,      // [Cin,Cout]
    const float* __restrict__ Bc,     // [Cout] conv bias
    const float* __restrict__ Gc,     // [Cout] bn gamma
    const float* __restrict__ BBc,    // [Cout] bn beta
    float* __restrict__ out,          // [B,S,Cout]
    int N, int S, int Cfeat, int Cout, int useXyz)
{
    const int ns   = MT * 16;
    const int s    = blockIdx.x;
    const int b    = blockIdx.y;
    const int tid  = threadIdx.x;
    const int wave = tid >> 5;
    const int lane = tid & 31;
    const int half = lane >> 4;
    const int l16  = lane & 15;
    const int coff = useXyz ? 3 : 0;
    const int Cin  = Cfeat + coff;

    __shared__ float Xs[MT * 16][KC];
    __shared__ int   ridx[MT * 16];
    __shared__ float sxyz[3];

    for (int e = tid; e < ns; e += 256)
        ridx[e] = idx[((size_t)b * S + s) * ns + e];
    if (useXyz && tid < 3)
        sxyz[tid] = nxyz[((size_t)b * S + s) * 3 + tid];
    __syncthreads();

    const int  ntile = blockIdx.z * 8 + wave;
    const int  n0    = ntile * 16;
    const bool ntOK  = (n0 < Cout);
    const int  ncol  = n0 + l16;

    v8f acc[MT];
    const v8f zero = {0.f, 0.f, 0.f, 0.f, 0.f, 0.f, 0.f, 0.f};
#pragma unroll
    for (int mt = 0; mt < MT; ++mt) acc[mt] = zero;

    for (int k0 = 0; k0 < Cin; k0 += KC) {
        // cooperative stage of X[:, k0:k0+KC] (zero padded past Cin)
        for (int e = tid; e < ns * KC; e += 256) {
            int row = e / KC;
            int kk  = e - row * KC;
            int k   = k0 + kk;
            float v = 0.0f;
            if (k < Cin) {
                int pid = ridx[row];
                if (useXyz && k < 3)
                    v = xyz[((size_t)b * N + pid) * 3 + k] - sxyz[k];
                else
                    v = feats[((size_t)b * N + pid) * Cfeat + (k - coff)];
            }
            Xs[row][kk] = v;
        }
        __syncthreads();

        if (ntOK) {
            int krem  = Cin - k0;
            int kkend = (krem >= KC) ? KC : ((krem + 3) & ~3);
            for (int kk = 0; kk < kkend; kk += 4) {
                int   kb = k0 + kk + 2 * half;
                float b0 = (kb     < Cin) ? W[(size_t)kb * Cout + ncol]       : 0.0f;
                float b1 = (kb + 1 < Cin) ? W[(size_t)(kb + 1) * Cout + ncol] : 0.0f;
                v2f Bf = {b0, b1};
#pragma unroll
                for (int mt = 0; mt < MT; ++mt) {
                    int r = mt * 16 + l16;
                    v2f Af = {Xs[r][kk + 2 * half], Xs[r][kk + 2 * half + 1]};
                    acc[mt] = __builtin_amdgcn_wmma_f32_16x16x4_f32(
                        false, Af, false, Bf, (short)0, acc[mt], false, false);
                }
            }
        }
        __syncthreads();
    }

    if (ntOK) {
        float bn = Bc[ncol], gn = Gc[ncol], bbn = BBc[ncol];
        float m = 0.0f; // relu output >= 0
#pragma unroll
        for (int mt = 0; mt < MT; ++mt) {
#pragma unroll
            for (int i = 0; i < 8; ++i) {
                float h = relu_f(gn * (acc[mt][i] + bn) * BN_SCALE + bbn);
                m = fmaxf(m, h);
            }
        }
        float other = __shfl_xor(m, 16, 32);
        m = fmaxf(m, other);
        if (lane < 16)
            out[((size_t)b * S + s) * Cout + ncol] = m;
    }
}

// ---------------------------------------------------------------------------
// fcat[b,s,:] = concat(xyz4(3), ds1(64), ds2(128), ds3(256), feats4(512))
// ---------------------------------------------------------------------------
__global__ __launch_bounds__(256) void build_fcat_kernel(
    const float* __restrict__ xyz4, const float* __restrict__ d1,
    const float* __restrict__ d2,   const float* __restrict__ d3,
    const float* __restrict__ f4,   float* __restrict__ fcat, int total)
{
    int e = blockIdx.x * 256 + threadIdx.x;
    if (e >= total) return;
    int c  = e % 963;
    int bs = e / 963;
    float v;
    if (c < 3)        v = xyz4[(size_t)bs * 3 + c];
    else if (c < 67)  v = d1[(size_t)bs * 64  + (c - 3)];
    else if (c < 195) v = d2[(size_t)bs * 128 + (c - 67)];
    else if (c < 451) v = d3[(size_t)bs * 256 + (c - 195)];
    else              v = f4[(size_t)bs * 512 + (c - 451)];
    fcat[e] = v;
}

__global__ __launch_bounds__(256) void iota_kernel(int* __restrict__ idx)
{
    int e = blockIdx.x * 256 + threadIdx.x; // 8 blocks -> b = blockIdx.x
    idx[e] = threadIdx.x;                   // identity row index per batch
}

// ---------------------------------------------------------------------------
// FC layers (tiny)
// ---------------------------------------------------------------------------
__global__ __launch_bounds__(256) void fc1_kernel(const float* __restrict__ g,
                                                  const float* __restrict__ W,
                                                  const float* __restrict__ bv,
                                                  const float* __restrict__ gv,
                                                  const float* __restrict__ bbv,
                                                  float* __restrict__ med)
{
    int e = blockIdx.x * 256 + threadIdx.x;
    if (e >= 8 * 512) return;
    int b = e / 512, o = e % 512;
    float acc = 0.0f;
    for (int k = 0; k < 1024; ++k)
        acc += g[(size_t)b * 1024 + k] * W[(size_t)k * 512 + o];
    acc += bv[o];
    med[e] = relu_f(gv[o] * acc * BN_SCALE + bbv[o]);
}

__global__ __launch_bounds__(256) void fc2_kernel(const float* __restrict__ med,
                                                  const float* __restrict__ W,
                                                  const float* __restrict__ bv,
                                                  float* __restrict__ pred)
{
    int e = blockIdx.x * 256 + threadIdx.x;
    if (e >= 8 * 40) return;
    int b = e / 40, o = e % 40;
    float acc = 0.0f;
    for (int k = 0; k < 512; ++k)
        acc += med[(size_t)b * 512 + k] * W[(size_t)k * 40 + o];
    pred[e] = acc + bv[o];
}

// ---------------------------------------------------------------------------
// Host orchestration
// ---------------------------------------------------------------------------
namespace {

struct LayerP { const float *w, *b, *g, *bb; };

static int find_size_idx(const int* sizes, int n, int v)
{
    for (int i = 0; i < n; ++i)
        if (sizes[i] == v) return i;
    return -1;
}

// Locate (w,b,g,bb) for a layer whose w has a unique flat size.
// Handles both insertion-order (w,b,g,bb) and sorted-key (b,bb,g,w) layouts.
static LayerP resolve_layer(void* const* d_in, const int* sizes, int n,
                            int wsize, int cout)
{
    LayerP p = {nullptr, nullptr, nullptr, nullptr};
    int wi = find_size_idx(sizes, n, wsize);
    if (wi < 0) return p;
    p.w = (const float*)d_in[wi];
    if (wi + 3 < n && sizes[wi + 1] == cout && sizes[wi + 2] == cout &&
        sizes[wi + 3] == cout) {
        p.b  = (const float*)d_in[wi + 1];
        p.g  = (const float*)d_in[wi + 2];
        p.bb = (const float*)d_in[wi + 3];
    } else if (wi >= 3 && sizes[wi - 1] == cout && sizes[wi - 2] == cout &&
               sizes[wi - 3] == cout) {
        p.b  = (const float*)d_in[wi - 3];
        p.bb = (const float*)d_in[wi - 2];
        p.g  = (const float*)d_in[wi - 1];
    }
    return p;
}

struct Scratch {
    float *xyz1, *xyz2, *xyz3, *xyz4;
    float *f1, *f2, *f3, *f4;
    float *d1, *d2, *d3;
    float *fcat, *gvec, *nxds;
    int   *fidx, *ball, *iota;
    size_t bytes;
};

static Scratch layout_ws(void* ws)
{
    char*  base = (char*)ws;
    size_t off  = 0;
    auto alloc = [&](size_t bytes) -> void* {
        void* p = base + off;
        off = (off + bytes + 255) & ~(size_t)255;
        return p;
    };
    Scratch s;
    s.xyz1 = (float*)alloc(8 * 2048 * 3 * 4);
    s.xyz2 = (float*)alloc(8 * 1024 * 3 * 4);
    s.xyz3 = (float*)alloc(8 * 512 * 3 * 4);
    s.xyz4 = (float*)alloc(8 * 256 * 3 * 4);
    s.f1   = (float*)alloc((size_t)8 * 2048 * 64 * 4);
    s.f2   = (float*)alloc((size_t)8 * 1024 * 128 * 4);
    s.f3   = (float*)alloc((size_t)8 * 512 * 256 * 4);
    s.f4   = (float*)alloc((size_t)8 * 256 * 512 * 4);
    s.d1   = (float*)alloc((size_t)8 * 256 * 64 * 4);
    s.d2   = (float*)alloc((size_t)8 * 256 * 128 * 4);
    s.d3   = (float*)alloc((size_t)8 * 256 * 256 * 4);
    s.fcat = (float*)alloc((size_t)8 * 256 * 963 * 4);
    s.gvec = (float*)alloc(8 * 1024 * 4);
    s.nxds = (float*)alloc(8 * 256 * 3 * 4);
    s.fidx = (int*)alloc(8 * 2048 * 4);
    s.ball = (int*)alloc((size_t)8 * 2048 * 32 * 4);
    s.iota = (int*)alloc(8 * 256 * 4);
    s.bytes = off;
    return s;
}

static inline int ceil_div(int a, int b) { return (a + b - 1) / b; }

struct Params {
    LayerP sa1, sa2, sa3, sa4, ds1, ds2, ds3, gsa, fc1;
    const float *fc2w, *fc2b;
};

static void run_branch(const float* pc, float* medOut, float* predOut,
                       const Params& P, const Scratch& S, hipStream_t st)
{
    // ---- sa1: N=4096 -> 2048, ns=32, Cin=3, Cout=64
    fps_kernel<<<8, 256, 0, st>>>(pc, 4096, 2048, S.fidx);
    gather_xyz_kernel<<<ceil_div(8 * 2048, 256), 256, 0, st>>>(pc, S.fidx, S.xyz1, 4096, 2048);
    ball_query_kernel<<<ceil_div(8 * 2048, 256), 256, 0, st>>>(pc, S.xyz1, S.ball, 4096, 2048, 32, 0.24f * 0.24f);
    sa_conv_kernel<2, 64><<<dim3(2048, 8, 1), 256, 0, st>>>(
        pc, nullptr, S.ball, S.xyz1, P.sa1.w, P.sa1.b, P.sa1.g, P.sa1.bb,
        S.f1, 4096, 2048, 0, 64, 1);

    // ---- sa2: N=2048 -> 1024, ns=48, Cin=67, Cout=128
    fps_kernel<<<8, 256, 0, st>>>(S.xyz1, 2048, 1024, S.fidx);
    gather_xyz_kernel<<<ceil_div(8 * 1024, 256), 256, 0, st>>>(S.xyz1, S.fidx, S.xyz2, 2048, 1024);
    ball_query_kernel<<<ceil_div(8 * 1024, 256), 256, 0, st>>>(S.xyz1, S.xyz2, S.ball, 2048, 1024, 48, 0.28f * 0.28f);
    sa_conv_kernel<3, 64><<<dim3(1024, 8, 1), 256, 0, st>>>(
        S.xyz1, S.f1, S.ball, S.xyz2, P.sa2.w, P.sa2.b, P.sa2.g, P.sa2.bb,
        S.f2, 2048, 1024, 64, 128, 1);

    // ---- sa3: N=1024 -> 512, ns=64, Cin=131, Cout=256
    fps_kernel<<<8, 256, 0, st>>>(S.xyz2, 1024, 512, S.fidx);
    gather_xyz_kernel<<<ceil_div(8 * 512, 256), 256, 0, st>>>(S.xyz2, S.fidx, S.xyz3, 1024, 512);
    ball_query_kernel<<<ceil_div(8 * 512, 256), 256, 0, st>>>(S.xyz2, S.xyz3, S.ball, 1024, 512, 64, 0.32f * 0.32f);
    sa_conv_kernel<4, 64><<<dim3(512, 8, 2), 256, 0, st>>>(
        S.xyz2, S.f2, S.ball, S.xyz3, P.sa3.w, P.sa3.b, P.sa3.g, P.sa3.bb,
        S.f3, 1024, 512, 128, 256, 1);

    // ---- sa4: N=512 -> 256, ns=64, Cin=259, Cout=512
    fps_kernel<<<8, 256, 0, st>>>(S.xyz3, 512, 256, S.fidx);
    gather_xyz_kernel<<<ceil_div(8 * 256, 256), 256, 0, st>>>(S.xyz3, S.fidx, S.xyz4, 512, 256);
    ball_query_kernel<<<ceil_div(8 * 256, 256), 256, 0, st>>>(S.xyz3, S.xyz4, S.ball, 512, 256, 64, 0.34f * 0.34f);
    sa_conv_kernel<4, 64><<<dim3(256, 8, 4), 256, 0, st>>>(
        S.xyz3, S.f3, S.ball, S.xyz4, P.sa4.w, P.sa4.b, P.sa4.g, P.sa4.bb,
        S.f4, 512, 256, 256, 512, 1);

    // ---- ds1: on (xyz1, f1): N=2048 -> 256, ns=64, Cin=64, Cout=64 (no xyz)
    fps_kernel<<<8, 256, 0, st>>>(S.xyz1, 2048, 256, S.fidx);
    gather_xyz_kernel<<<ceil_div(8 * 256, 256), 256, 0, st>>>(S.xyz1, S.fidx, S.nxds, 2048, 256);
    ball_query_kernel<<<ceil_div(8 * 256, 256), 256, 0, st>>>(S.xyz1, S.nxds, S.ball, 2048, 256, 64, 0.32f * 0.32f);
    sa_conv_kernel<4, 64><<<dim3(256, 8, 1), 256, 0, st>>>(
        nullptr, S.f1, S.ball, nullptr, P.ds1.w, P.ds1.b, P.ds1.g, P.ds1.bb,
        S.d1, 2048, 256, 64, 64, 0);

    // ---- ds2: on (xyz2, f2): N=1024 -> 256, Cin=128, Cout=128
    fps_kernel<<<8, 256, 0, st>>>(S.xyz2, 1024, 256, S.fidx);
    gather_xyz_kernel<<<ceil_div(8 * 256, 256), 256, 0, st>>>(S.xyz2, S.fidx, S.nxds, 1024, 256);
    ball_query_kernel<<<ceil_div(8 * 256, 256), 256, 0, st>>>(S.xyz2, S.nxds, S.ball, 1024, 256, 64, 0.32f * 0.32f);
    sa_conv_kernel<4, 64><<<dim3(256, 8, 1), 256, 0, st>>>(
        nullptr, S.f2, S.ball, nullptr, P.ds2.w, P.ds2.b, P.ds2.g, P.ds2.bb,
        S.d2, 1024, 256, 128, 128, 0);

    // ---- ds3: on (xyz3, f3): N=512 -> 256, Cin=256, Cout=256
    fps_kernel<<<8, 256, 0, st>>>(S.xyz3, 512, 256, S.fidx);
    gather_xyz_kernel<<<ceil_div(8 * 256, 256), 256, 0, st>>>(S.xyz3, S.fidx, S.nxds, 512, 256);
    ball_query_kernel<<<ceil_div(8 * 256, 256), 256, 0, st>>>(S.xyz3, S.nxds, S.ball, 512, 256, 64, 0.32f * 0.32f);
    sa_conv_kernel<4, 64><<<dim3(256, 8, 2), 256, 0, st>>>(
        nullptr, S.f3, S.ball, nullptr, P.ds3.w, P.ds3.b, P.ds3.g, P.ds3.bb,
        S.d3, 512, 256, 256, 256, 0);

    // ---- global SA: concat -> [8,256,963] @ [963,1024] + maxpool over 256
    int tot = 8 * 256 * 963;
    build_fcat_kernel<<<ceil_div(tot, 256), 256, 0, st>>>(S.xyz4, S.d1, S.d2, S.d3, S.f4, S.fcat, tot);
    iota_kernel<<<8, 256, 0, st>>>(S.iota);
    sa_conv_kernel<16, 16><<<dim3(1, 8, 8), 256, 0, st>>>(
        nullptr, S.fcat, S.iota, nullptr, P.gsa.w, P.gsa.b, P.gsa.g, P.gsa.bb,
        S.gvec, 256, 1, 963, 1024, 0);

    // ---- FC head
    fc1_kernel<<<ceil_div(8 * 512, 256), 256, 0, st>>>(S.gvec, P.fc1.w, P.fc1.b, P.fc1.g, P.fc1.bb, medOut);
    fc2_kernel<<<ceil_div(8 * 40, 256), 256, 0, st>>>(medOut, P.fc2w, P.fc2b, predOut);
}

} // namespace

extern "C" void kernel_launch(void* const* d_in, const int* in_sizes, int n_in,
                              void* d_out, int out_size, void* d_ws, size_t ws_size,
                              hipStream_t stream)
{
    // ---- resolve point clouds (flat size 8*4096*3 = 98304)
    const float *pcA = nullptr, *pcB = nullptr;
    for (int i = 0; i < n_in; ++i) {
        if (in_sizes[i] == 98304) {
            if (!pcA) pcA = (const float*)d_in[i];
            else if (!pcB) { pcB = (const float*)d_in[i]; break; }
        }
    }
    if (!pcA || !pcB) return;

    // ---- resolve params by unique weight sizes
    Params P;
    P.sa1 = resolve_layer(d_in, in_sizes, n_in, 3 * 64, 64);
    P.sa2 = resolve_layer(d_in, in_sizes, n_in, 67 * 128, 128);
    P.sa3 = resolve_layer(d_in, in_sizes, n_in, 131 * 256, 256);
    P.sa4 = resolve_layer(d_in, in_sizes, n_in, 259 * 512, 512);
    P.ds1 = resolve_layer(d_in, in_sizes, n_in, 64 * 64, 64);
    P.ds2 = resolve_layer(d_in, in_sizes, n_in, 128 * 128, 128);
    P.ds3 = resolve_layer(d_in, in_sizes, n_in, 256 * 256, 256);
    P.gsa = resolve_layer(d_in, in_sizes, n_in, 963 * 1024, 1024);
    P.fc1 = resolve_layer(d_in, in_sizes, n_in, 1024 * 512, 512);
    {
        int wi = find_size_idx(in_sizes, n_in, 512 * 40);
        if (wi < 0) return;
        P.fc2w = (const float*)d_in[wi];
        if (wi + 1 < n_in && in_sizes[wi + 1] == 40)
            P.fc2b = (const float*)d_in[wi + 1];
        else if (wi >= 1 && in_sizes[wi - 1] == 40)
            P.fc2b = (const float*)d_in[wi - 1];
        else
            return;
    }
    if (!P.sa1.b || !P.sa2.b || !P.sa3.b || !P.sa4.b || !P.ds1.b || !P.ds2.b ||
        !P.ds3.b || !P.gsa.b || !P.fc1.b)
        return;

    Scratch S = layout_ws(d_ws);
    if (S.bytes > ws_size) return;

    float* out = (float*)d_out;
    float* medA  = out;                 // [8,512]
    float* predA = out + 8 * 512;       // [8,40]
    float* medB  = predA + 8 * 40;      // [8,512]
    float* predB = medB + 8 * 512;      // [8,40]
    (void)out_size;

    run_branch(pcA, medA, predA, P, S, stream);
    run_branch(pcB, medB, predB, P, S, stream);
}